// Dcls2dFull_44487271252352
// MI455X (gfx1250) — compile-verified
//
#include <hip/hip_runtime.h>
#include <hip/hip_bf16.h>

// ---------------------------------------------------------------------------
// Dcls2dFull forward for MI455X (gfx1250, wave32, WMMA + TDM)
//   out[32,64,95,95] = conv2d(x[32,64,96,96], dense_kernel[64,64,12,12], pad=5)
// Implicit-GEMM with V_WMMA_F32_16X16X4_F32 (exact f32 matrix FMA).
// Per-tap kernel slices double-buffered in LDS via TENSOR_LOAD_TO_LDS (TDM),
// software-pipelined against the WMMA loop (1 barrier / tap, DMA overlapped).
// ---------------------------------------------------------------------------

typedef __attribute__((ext_vector_type(2))) float v2f;
typedef __attribute__((ext_vector_type(8))) float v8f;
typedef __attribute__((ext_vector_type(4), aligned(4))) float v4fa;  // unaligned-ok b128
typedef __attribute__((ext_vector_type(4))) unsigned int u32x4;
typedef __attribute__((ext_vector_type(8))) int i32x8;
typedef __attribute__((ext_vector_type(4))) int i32x4;

#if __has_builtin(__builtin_amdgcn_tensor_load_to_lds) && \
    __has_builtin(__builtin_amdgcn_s_wait_tensorcnt)
#define USE_TDM 1
#endif

namespace {
constexpr int NB   = 32;   // batch
constexpr int CIN  = 64;
constexpr int COUT = 64;
constexpr int H    = 96, W = 96;
constexpr int OH   = 95, OW = 95;
constexpr int LIM  = 12;          // dense kernel extent
constexpr int PADC = 5;
constexpr int NTAP = LIM * LIM;   // 144
constexpr int TX   = 16;          // output pixels per tile (x)
constexpr int TY   = 4;           // output rows per tile (y)
constexpr int NTX  = (OW + TX - 1) / TX;  // 6
constexpr int NTY  = (OH + TY - 1) / TY;  // 24
constexpr int KWELEM = NTAP * COUT * CIN; // 589824 floats in workspace
constexpr int QCIN  = 16;                 // cin per pass
constexpr int NPASS = CIN / QCIN;         // 4
// LDS input tile: 16 cin x 15 rows x 28-col pitch (27 valid)
constexpr int XROWS  = TY + LIM - 1;      // 15
constexpr int XCOLS  = TX + LIM - 1;      // 27 valid
constexpr int XPITCH = 28;                // padded pitch
constexpr int XCIN   = XROWS * XPITCH;    // 420 floats per cin plane
}

// --- step 0: zero the dense-kernel workspace (ws is not re-poisoned) --------
__global__ void dcls_zero_ws(float* __restrict__ wk, int n) {
    int i = blockIdx.x * blockDim.x + threadIdx.x;
    if (i < n) wk[i] = 0.0f;
}

// --- step 1: bilinear scatter of 3x3 weights into dense 12x12 kernel -------
// Layout: wk[(th*12+tw)*4096 + cout*64 + cin]  (per-tap [cout][cin] slice)
// One thread per (cout,cin): owns all its 36 scattered adds -> no atomics,
// fully deterministic.
__global__ void dcls_build_kernel(const float* __restrict__ w,
                                  const float* __restrict__ P1,
                                  const float* __restrict__ P2,
                                  float* __restrict__ wk) {
    int gid = blockIdx.x * blockDim.x + threadIdx.x;   // (co,ci)
    if (gid >= COUT * CIN) return;
    int base = gid;                                    // co*64 + ci
    #pragma unroll
    for (int k = 0; k < 9; ++k) {
        int wi = gid * 9 + k;                          // (co,ci,kh,kw) flat
        float wv = w[wi];
        float p1 = fminf(fmaxf(P1[wi] + (float)(LIM / 2), 0.0f), (float)(LIM - 1));
        float p2 = fminf(fmaxf(P2[wi] + (float)(LIM / 2), 0.0f), (float)(LIM - 1));
        float f1 = floorf(p1), f2 = floorf(p2);
        int   i1 = (int)f1,   i2 = (int)f2;
        float r1 = p1 - f1,   r2 = p2 - f2;
        int   i1p = min(i1 + 1, LIM - 1);
        int   i2p = min(i2 + 1, LIM - 1);
        wk[(i1  * LIM + i2 ) * (COUT * CIN) + base] += wv * (1.0f - r1) * (1.0f - r2);
        wk[(i1p * LIM + i2 ) * (COUT * CIN) + base] += wv * r1 * (1.0f - r2);
        wk[(i1  * LIM + i2p) * (COUT * CIN) + base] += wv * (1.0f - r1) * r2;
        wk[(i1p * LIM + i2p) * (COUT * CIN) + base] += wv * r1 * r2;
    }
}

#if defined(USE_TDM)
// Issue one TDM 2-D tile load: 64 rows x QCIN f32, row stride CIN, into LDS.
__device__ __forceinline__ void tdm_load_slice(const float* gsrc, unsigned ldsOff) {
    const unsigned long long ga = (unsigned long long)(uintptr_t)gsrc;
    u32x4 g0;
    g0.x = 1u;                                   // count=1, user mode
    g0.y = ldsOff;                               // lds_addr      [63:32]
    g0.z = (unsigned)(ga & 0xffffffffu);         // global_addr lo [95:64]
    g0.w = (unsigned)((ga >> 32) & 0x1ffffffu)   // global_addr hi [120:96]
         | (2u << 30);                           // type=2 (image) [127:126]
    i32x8 g1;
    g1[0] = 0x00020000;          // wg_mask=0, data_size=2 (4 bytes)
    g1[1] = (int)(64u << 16);    // tensor_dim0 = 64 (lo16 @ [63:48])
    g1[2] = (int)(9216u << 16);  // dim0 hi=0, tensor_dim1 = 9216 (lo16)
    g1[3] = (int)((unsigned)QCIN << 16);  // dim1 hi=0, tile_dim0 = 16
    g1[4] = 64;                  // tile_dim1 = 64, tile_dim2 = 0
    g1[5] = CIN;                 // tensor_dim0_stride = 64 (lo32)
    g1[6] = 0;                   // stride0 hi, stride1 lo
    g1[7] = 0;                   // stride1 hi
    i32x4 z4 = {0, 0, 0, 0};                 // groups 2/3 unused (2-D tile)
    i32x8 z8 = {0, 0, 0, 0, 0, 0, 0, 0};
    __builtin_amdgcn_tensor_load_to_lds(g0, g1, z4, z4, z8, 0);
}
#endif

// --- step 2: implicit-GEMM conv with f32 WMMA -------------------------------
// Block = 128 threads (4 waves). Wave wv owns couts [wv*16, wv*16+16).
// Block computes out rows [yb, yb+4) x cols [xb, xb+16) x all 64 couts.
__global__ __launch_bounds__(128) void dcls_conv_wmma(
        const float* __restrict__ x,
        const float* __restrict__ wk,
        const float* __restrict__ bias,
        float* __restrict__ out) {
    __shared__ float xs[QCIN * XCIN];        // 6720 f32 = 26.25 KB
    __shared__ float Bs[2][COUT * QCIN];     // 2 x 4 KB (double-buffered slice)

    const int tid  = threadIdx.x;
    const int lane = tid & 31;
    const int wv   = tid >> 5;
    const int hi   = lane >> 4;     // 0: K pair {0,1}, 1: K pair {2,3}
    const int m    = lane & 15;     // A: M (pixel) index; B/C: N (cout) index
    const int xb   = blockIdx.x * TX;
    const int yb   = blockIdx.y * TY;
    const int n    = blockIdx.z;
    const int co   = wv * 16 + m;   // this lane's cout column in B/C

    v8f zero8 = {0.f, 0.f, 0.f, 0.f, 0.f, 0.f, 0.f, 0.f};
    v8f acc[TY];
    #pragma unroll
    for (int mt = 0; mt < TY; ++mt) acc[mt] = zero8;

#if defined(USE_TDM)
    const unsigned ldsBs0 = (unsigned)(uintptr_t)(&Bs[0][0]);
    const unsigned ldsBs1 = (unsigned)(uintptr_t)(&Bs[1][0]);
#endif

    #pragma unroll 1
    for (int pass = 0; pass < NPASS; ++pass) {
        __syncthreads();   // protect xs & Bs[0] from previous pass's readers
        // ---- cooperative load of input halo tile (zero-padded borders) ----
        #pragma unroll 1
        for (int idx = tid; idx < QCIN * XROWS * XCOLS; idx += 128) {
            int ciL = idx / (XROWS * XCOLS);
            int rem = idx - ciL * (XROWS * XCOLS);
            int r   = rem / XCOLS;
            int cc  = rem - r * XCOLS;
            int ir  = yb - PADC + r;
            int ic  = xb - PADC + cc;
            float v = 0.0f;
            if (ir >= 0 && ir < H && ic >= 0 && ic < W)
                v = x[(((size_t)n * CIN + (pass * QCIN + ciL)) * H + ir) * W + ic];
            xs[ciL * XCIN + r * XPITCH + cc] = v;
        }

        // ---- prologue: start DMA of tap 0 slice into buffer 0 --------------
#if defined(USE_TDM)
        if (wv == 0)
            tdm_load_slice(wk + pass * QCIN, ldsBs0);
#else
        #pragma unroll 1
        for (int idx = tid; idx < COUT * QCIN; idx += 128)
            Bs[0][idx] = wk[(idx >> 4) * CIN + pass * QCIN + (idx & 15)];
#endif

        int th = 0, tw = 0;
        #pragma unroll 1
        for (int tap = 0; tap < NTAP; ++tap) {
#if defined(USE_TDM)
            if (wv == 0) __builtin_amdgcn_s_wait_tensorcnt(0);
#endif
            __syncthreads();   // publishes xs (tap 0) + slice buffer tap&1;
                               // also means buffer (tap+1)&1 is free again
            // ---- issue next slice DMA into the other buffer ----------------
            if (tap + 1 < NTAP) {
#if defined(USE_TDM)
                if (wv == 0)
                    tdm_load_slice(wk + (size_t)(tap + 1) * (COUT * CIN) + pass * QCIN,
                                   (tap & 1) ? ldsBs0 : ldsBs1);
#else
                float* nb = &Bs[(tap + 1) & 1][0];
                #pragma unroll 1
                for (int idx = tid; idx < COUT * QCIN; idx += 128)
                    nb[idx] = wk[(size_t)(tap + 1) * (COUT * CIN)
                                 + (idx >> 4) * CIN + pass * QCIN + (idx & 15)];
#endif
            }

            // ---- 16 WMMAs, all LDS offsets are immediates ------------------
            const float* ap = &xs[th * XPITCH + tw + m + hi * (2 * XCIN)];
            const float* bp = &Bs[tap & 1][co * QCIN + 2 * hi];
            #pragma unroll
            for (int kc = 0; kc < QCIN / 4; ++kc) {
                v2f b = *(const v2f*)(bp + kc * 4);          // ds_load_b64
                #pragma unroll
                for (int mt = 0; mt < TY; ++mt) {
                    v2f a = { ap[kc * (4 * XCIN) + mt * XPITCH],
                              ap[kc * (4 * XCIN) + mt * XPITCH + XCIN] };
                    acc[mt] = __builtin_amdgcn_wmma_f32_16x16x4_f32(
                        false, a, false, b, (short)0, acc[mt], false, false);
                }
            }
            if (++tw == LIM) { tw = 0; ++th; }
        }
    }

    // ---- epilogue: C layout -> global stores (+bias) ----
    const float bv = bias[co];
    float* const orow0 = out + ((size_t)n * COUT + co) * OH * OW;
    if (xb + TX <= OW) {
        // interior in x: two b128 stores per row (rows are 95 f32 -> align(4))
        #pragma unroll
        for (int mt = 0; mt < TY; ++mt) {
            const int y = yb + mt;
            if (y < OH) {
                float* p = orow0 + (size_t)y * OW + xb + 8 * hi;
                v4fa lo = { acc[mt][0] + bv, acc[mt][1] + bv,
                            acc[mt][2] + bv, acc[mt][3] + bv };
                v4fa hv = { acc[mt][4] + bv, acc[mt][5] + bv,
                            acc[mt][6] + bv, acc[mt][7] + bv };
                *(v4fa*)(p)     = lo;
                *(v4fa*)(p + 4) = hv;
            }
        }
    } else {
        #pragma unroll
        for (int mt = 0; mt < TY; ++mt) {
            const int y = yb + mt;
            if (y < OH) {
                #pragma unroll
                for (int v = 0; v < 8; ++v) {
                    const int px = xb + v + 8 * hi;   // C: VGPR v, lane-half -> M
                    if (px < OW)
                        orow0[(size_t)y * OW + px] = acc[mt][v] + bv;
                }
            }
        }
    }
}

// ---------------------------------------------------------------------------
extern "C" void kernel_launch(void* const* d_in, const int* in_sizes, int n_in,
                              void* d_out, int out_size, void* d_ws, size_t ws_size,
                              hipStream_t stream) {
    const float* x    = (const float*)d_in[0];
    const float* w    = (const float*)d_in[1];
    const float* P1   = (const float*)d_in[2];
    const float* P2   = (const float*)d_in[3];
    const float* bias = (const float*)d_in[4];
    float* out = (float*)d_out;
    float* wk  = (float*)d_ws;   // 589824 f32 = 2.36 MB dense kernel

    dcls_zero_ws<<<(KWELEM + 255) / 256, 256, 0, stream>>>(wk, KWELEM);
    dcls_build_kernel<<<(COUT * CIN + 255) / 256, 256, 0, stream>>>(w, P1, P2, wk);

    dim3 grid(NTX, NTY, NB);   // 6 x 24 x 32
    dcls_conv_wmma<<<grid, 128, 0, stream>>>(x, wk, bias, out);
}